// Stickbreaking_67542655696927
// MI455X (gfx1250) — compile-verified
//
#include <hip/hip_runtime.h>
#include <hip/hip_bf16.h>

typedef __attribute__((ext_vector_type(2))) float v2f;
typedef __attribute__((ext_vector_type(8))) float v8f;

#define S_LEN 1024
#define HD    64
#define NW    4          // waves per workgroup
#define WROW  18         // padded LDS row stride (even for b64 align, conflict-free)

__device__ __forceinline__ v8f wmma_f32(v2f a, v2f b, v8f c) {
  // D = A(16x4 f32) * B(4x16 f32) + C(16x16 f32)
  return __builtin_amdgcn_wmma_f32_16x16x4_f32(false, a, false, b, (short)0, c,
                                               false, false);
}

__global__ __launch_bounds__(128) void stickbreak_kernel(
    const float* __restrict__ q, const float* __restrict__ k,
    const float* __restrict__ v, float* __restrict__ out) {
  __shared__ float lds[NW * 16 * WROW];

  const int bh   = blockIdx.x;            // b*H + h
  const int wave = threadIdx.x >> 5;
  const int lane = threadIdx.x & 31;
  const int m    = lane & 15;             // column-in-tile / row-for-A
  const int hi   = lane >> 4;             // wave half

  const int qt = (blockIdx.y * NW + wave) * 16;   // query tile base row

  const size_t base = (size_t)bh * S_LEN * HD;
  const float* qb = q + base;
  const float* kb = k + base;
  const float* vb = v + base;
  float*       ob = out + base;

  const float scale = 0.125f;             // 1/sqrt(64)

  // ---- Q tile in WMMA-A layout, pre-scaled. qa[c] covers d = 4c..4c+3 ----
  v2f qa[16];
#pragma unroll
  for (int c = 0; c < 16; ++c) {
    const float* p = qb + (size_t)(qt + m) * HD + 4 * c + 2 * hi;
    v2f t = *(const v2f*)p;
    qa[c] = t * scale;
  }

  v8f acc0 = {}, acc1 = {}, acc2 = {}, acc3 = {};
  float carry[8];
#pragma unroll
  for (int r = 0; r < 8; ++r) carry[r] = 0.0f;

  float* wl = lds + wave * 16 * WROW;     // wave-private 16x16 tile

  for (int j0 = qt; j0 >= 0; j0 -= 16) {
    if (j0 >= 16) {   // prefetch next key block (descending j)
      __builtin_prefetch(kb + (size_t)(j0 - 16 + m) * HD, 0, 3);
      __builtin_prefetch(vb + (size_t)(j0 - 16 + m) * HD, 0, 3);
    }

    // ---- logits tile: L = (Q*scale) @ K^T via f32 WMMA, accumulate D=64 ----
    v8f L = {};
#pragma unroll
    for (int c = 0; c < 16; ++c) {
      const float* p = kb + (size_t)(j0 + m) * HD + 4 * c + 2 * hi;
      v2f kt = *(const v2f*)p;            // B-layout: {d, d+1} = vgpr0, vgpr1
      L = wmma_f32(qa[c], kt, L);
    }

    const bool diag = (j0 == qt);

    // ---- z = sigmoid(L), lb = logsigmoid(-L), masked; suffix scan over j ----
    float w[8];
#pragma unroll
    for (int r = 0; r < 8; ++r) {
      float x  = L[r];
      float zz = 1.0f / (1.0f + __expf(-x));
      float sp = fmaxf(x, 0.0f) + __logf(1.0f + __expf(-fabsf(x)));
      float lb = -sp;
      if (diag && (m > r + 8 * hi)) { zz = 0.0f; lb = 0.0f; }  // j > i
      // inclusive suffix sum across the 16-lane half (j dimension)
      float s = lb;
#pragma unroll
      for (int off = 1; off < 16; off <<= 1) {
        float t = __shfl(s, lane + off);
        if (m + off < 16) s += t;
      }
      float tot = __shfl(s, hi << 4);     // block total for this row
      w[r] = zz * __expf(s + carry[r]);   // exp arg <= 0, no overflow
      carry[r] += tot;
    }

    // ---- bounce W through LDS: C-layout -> A-layout ----
#pragma unroll
    for (int r = 0; r < 8; ++r)
      wl[(r + 8 * hi) * WROW + m] = w[r];
    asm volatile("s_wait_dscnt 0x0" ::: "memory");

    v2f wa[4];
#pragma unroll
    for (int c = 0; c < 4; ++c)
      wa[c] = *(const v2f*)&wl[m * WROW + 4 * c + 2 * hi];

    // ---- out += W(16x16) @ V(16x64), 4 K-chunks x 4 N-chunks ----
#pragma unroll
    for (int c = 0; c < 4; ++c) {
      const float* vr0 = vb + (size_t)(j0 + 4 * c + 2 * hi) * HD + m;
      const float* vr1 = vr0 + HD;
      v2f b0 = { vr0[0],  vr1[0]  };
      v2f b1 = { vr0[16], vr1[16] };
      v2f b2 = { vr0[32], vr1[32] };
      v2f b3 = { vr0[48], vr1[48] };
      acc0 = wmma_f32(wa[c], b0, acc0);
      acc1 = wmma_f32(wa[c], b1, acc1);
      acc2 = wmma_f32(wa[c], b2, acc2);
      acc3 = wmma_f32(wa[c], b3, acc3);
    }
  }

  // ---- store out tile (C layout: lane = d-col, vgpr = query row) ----
#pragma unroll
  for (int r = 0; r < 8; ++r) {
    size_t row = (size_t)(qt + r + 8 * hi) * HD;
    ob[row + 0  + m] = acc0[r];
    ob[row + 16 + m] = acc1[r];
    ob[row + 32 + m] = acc2[r];
    ob[row + 48 + m] = acc3[r];
  }
}

extern "C" void kernel_launch(void* const* d_in, const int* in_sizes, int n_in,
                              void* d_out, int out_size, void* d_ws, size_t ws_size,
                              hipStream_t stream) {
  const float* q = (const float*)d_in[0];
  const float* k = (const float*)d_in[1];
  const float* v = (const float*)d_in[2];
  float* out = (float*)d_out;

  const int BH = in_sizes[0] / (S_LEN * HD);      // = 32
  dim3 grid(BH, S_LEN / (16 * NW));               // (32, 16)
  dim3 block(32 * NW);                            // 4 waves
  hipLaunchKernelGGL(stickbreak_kernel, grid, block, 0, stream, q, k, v, out);
}